// GCN_34754875359294
// MI455X (gfx1250) — compile-verified
//
#include <hip/hip_runtime.h>
#include <cstdint>
#include <cstddef>

typedef __attribute__((ext_vector_type(2))) float v2f;
typedef __attribute__((ext_vector_type(8))) float v8f;

// ---------------------------------------------------------------------------
// WMMA fp32 GEMM:  out[nrows x NCOL] = A[nrows x 64] @ W[64 x NCOL] (+bias)(+relu)
// Block = 256 threads = 8 waves; each wave owns a 16-row strip; block covers
// 128 rows. Uses V_WMMA_F32_16X16X4_F32 (full fp32, matches reference dtype).
// EPI: 0 = none, 1 = +bias, 2 = +bias +relu
// ---------------------------------------------------------------------------
template <int NCOL, int EPI>
__global__ __launch_bounds__(256) void gemm_wmma_kernel(
    const float* __restrict__ A, const float* __restrict__ W,
    const float* __restrict__ bias, float* __restrict__ out, int nrows)
{
    constexpr int NT = NCOL / 16;
    __shared__ __align__(16) float sW[64 * NCOL];
    __shared__ __align__(16) float sA[8][16 * 68];   // pitch 68 -> conflict-free

    const int tid  = threadIdx.x;
    const int wave = tid >> 5;
    const int lane = tid & 31;
    const int hh   = lane >> 4;   // half-wave select
    const int l16  = lane & 15;
    const int r0   = blockIdx.x * 128 + wave * 16;

    // Stage W (64 x NCOL, row-major) in LDS, shared by all waves.
    for (int i = tid; i < 64 * NCOL; i += 256) sW[i] = W[i];

    // Stage this wave's 16x64 A strip (coalesced float4 loads).
    const float4* A4 = reinterpret_cast<const float4*>(A);
    #pragma unroll
    for (int i = 0; i < 8; ++i) {
        int q  = lane + 32 * i;   // 0..255
        int r  = q >> 4;          // 0..15 row in strip
        int c4 = q & 15;          // 0..15 float4 column
        int gr = r0 + r;
        float4 v = make_float4(0.f, 0.f, 0.f, 0.f);
        if (gr < nrows) v = A4[(size_t)gr * 16 + c4];
        *reinterpret_cast<float4*>(&sA[wave][r * 68 + c4 * 4]) = v;
    }
    __syncthreads();

    v8f acc[NT] = {};

    #pragma unroll
    for (int kb = 0; kb < 64; kb += 4) {
        // A 16x4 fragment: lanes 0-15 -> K = kb+{0,1}; lanes 16-31 -> kb+{2,3}
        v2f a;
        a.x = sA[wave][l16 * 68 + kb + 2 * hh + 0];
        a.y = sA[wave][l16 * 68 + kb + 2 * hh + 1];
        #pragma unroll
        for (int nt = 0; nt < NT; ++nt) {
            // B 4x16 fragment: VGPR j holds row K = kb + j + 2*hh, col = l16
            v2f b;
            b.x = sW[(kb + 2 * hh + 0) * NCOL + nt * 16 + l16];
            b.y = sW[(kb + 2 * hh + 1) * NCOL + nt * 16 + l16];
            acc[nt] = __builtin_amdgcn_wmma_f32_16x16x4_f32(
                false, a, false, b, (short)0, acc[nt], false, false);
        }
    }

    // Epilogue: D layout -> row = r0 + j + 8*hh, col = nt*16 + l16
    #pragma unroll
    for (int nt = 0; nt < NT; ++nt) {
        const int col = nt * 16 + l16;
        float bv = 0.f;
        if (EPI > 0) bv = bias[col];
        #pragma unroll
        for (int j = 0; j < 8; ++j) {
            int gr = r0 + j + 8 * hh;
            if (gr < nrows) {
                float v = acc[nt][j] + bv;
                if (EPI == 2) v = fmaxf(v, 0.f);
                out[(size_t)gr * NCOL + col] = v;
            }
        }
    }
}

// ---------------------------------------------------------------------------
// Degree / normalization kernels
// ---------------------------------------------------------------------------
__global__ __launch_bounds__(256) void deg_init_kernel(float* __restrict__ deg, int n)
{
    int i = blockIdx.x * 256 + threadIdx.x;
    if (i < n) deg[i] = 1.0f;                 // self-loop contributes 1
}

__global__ __launch_bounds__(256) void deg_count_kernel(
    const int* __restrict__ col, float* __restrict__ deg, int ne)
{
    int e = blockIdx.x * 256 + threadIdx.x;
    if (e < ne) unsafeAtomicAdd(&deg[col[e]], 1.0f);
}

__global__ __launch_bounds__(256) void dinv_kernel(float* __restrict__ d, int n)
{
    int i = blockIdx.x * 256 + threadIdx.x;
    if (i < n) d[i] = rsqrtf(d[i]);           // deg >= 1 always
}

// agg[i,:] = dinv[i]^2 * xw[i,:]   (self-loop term seeds the accumulator)
__global__ __launch_bounds__(256) void self_term_kernel(
    const float* __restrict__ xw, const float* __restrict__ dinv,
    float* __restrict__ agg, int n)
{
    int t = blockIdx.x * 256 + threadIdx.x;
    if (t < n * 64) {
        int i = t >> 6;
        float dv = dinv[i];
        agg[t] = xw[t] * dv * dv;
    }
}

// One wave per edge: agg[col,:] += dinv[row]*dinv[col] * xw[row,:]
__global__ __launch_bounds__(256) void edge_scatter_kernel(
    const int* __restrict__ row, const int* __restrict__ col,
    const float* __restrict__ dinv, const float* __restrict__ xw,
    float* __restrict__ agg, int ne)
{
    int t    = blockIdx.x * 256 + threadIdx.x;
    int e    = t >> 5;
    int lane = t & 31;
    if (e >= ne) return;
    int   r  = row[e];
    int   c  = col[e];
    float nm = dinv[r] * dinv[c];
    const float* src = xw  + (size_t)r * 64;
    float*       dst = agg + (size_t)c * 64;
    unsafeAtomicAdd(&dst[lane],      nm * src[lane]);
    unsafeAtomicAdd(&dst[lane + 32], nm * src[lane + 32]);
}

// out = relu( (agg + b - mean) * rsqrt(var+eps) * gamma + beta )
__global__ __launch_bounds__(256) void bn_relu_kernel(
    const float* __restrict__ agg, const float* __restrict__ b,
    const float* __restrict__ g, const float* __restrict__ beta,
    const float* __restrict__ mean, const float* __restrict__ var,
    float* __restrict__ out, int n)
{
    int t = blockIdx.x * 256 + threadIdx.x;
    if (t < n * 64) {
        int f = t & 63;
        float v = (agg[t] + b[f] - mean[f]) * rsqrtf(var[f] + 1e-5f) * g[f] + beta[f];
        out[t] = fmaxf(v, 0.f);
    }
}

// ---------------------------------------------------------------------------
// Launch
// ---------------------------------------------------------------------------
extern "C" void kernel_launch(void* const* d_in, const int* in_sizes, int n_in,
                              void* d_out, int out_size, void* d_ws, size_t ws_size,
                              hipStream_t stream)
{
    const float* x      = (const float*)d_in[0];
    const int*   ei     = (const int*)d_in[1];
    const float* W0     = (const float*)d_in[2];
    const float* b0     = (const float*)d_in[3];
    const float* W1     = (const float*)d_in[4];
    const float* b1     = (const float*)d_in[5];
    const float* bn0_g  = (const float*)d_in[6];
    const float* bn0_b  = (const float*)d_in[7];
    const float* bn0_m  = (const float*)d_in[8];
    const float* bn0_v  = (const float*)d_in[9];
    const float* bn1_g  = (const float*)d_in[10];
    const float* bn1_b  = (const float*)d_in[11];
    const float* bn1_m  = (const float*)d_in[12];
    const float* bn1_v  = (const float*)d_in[13];
    const float* lin1_W = (const float*)d_in[14];
    const float* lin1_b = (const float*)d_in[15];
    const float* lin2_W = (const float*)d_in[16];
    const float* lin2_b = (const float*)d_in[17];
    const float* lin3_W = (const float*)d_in[18];
    const float* lin3_b = (const float*)d_in[19];

    const int N = in_sizes[0] / 64;   // 100000 nodes
    const int E = in_sizes[1] / 2;    // 1000000 edges
    const int* row = ei;
    const int* col = ei + E;

    // Workspace layout: dinv [N], bufA [N*64], bufB [N*64]
    char*  ws   = (char*)d_ws;
    size_t off0 = 0;
    size_t off1 = (off0 + (size_t)N * sizeof(float) + 255) & ~(size_t)255;
    size_t off2 = (off1 + (size_t)N * 64 * sizeof(float) + 255) & ~(size_t)255;
    float* dinv = (float*)(ws + off0);
    float* bufA = (float*)(ws + off1);
    float* bufB = (float*)(ws + off2);

    const dim3 blk(256);
    const int gN    = (N + 255) / 256;
    const int gE    = (E + 255) / 256;
    const int gNF   = (N * 64 + 255) / 256;
    const int gEdge = (E * 32 + 255) / 256;   // wave per edge
    const int gGemm = (N + 127) / 128;

    // --- symmetric normalization: dinv = rsqrt(1 + indeg) ---
    deg_init_kernel<<<gN, blk, 0, stream>>>(dinv, N);
    deg_count_kernel<<<gE, blk, 0, stream>>>(col, dinv, E);
    dinv_kernel<<<gN, blk, 0, stream>>>(dinv, N);

    // --- conv layer 0 ---
    gemm_wmma_kernel<64, 0><<<gGemm, blk, 0, stream>>>(x, W0, nullptr, bufA, N);
    self_term_kernel<<<gNF, blk, 0, stream>>>(bufA, dinv, bufB, N);
    edge_scatter_kernel<<<gEdge, blk, 0, stream>>>(row, col, dinv, bufA, bufB, E);
    bn_relu_kernel<<<gNF, blk, 0, stream>>>(bufB, b0, bn0_g, bn0_b, bn0_m, bn0_v, bufA, N);

    // --- conv layer 1 ---
    gemm_wmma_kernel<64, 0><<<gGemm, blk, 0, stream>>>(bufA, W1, nullptr, bufB, N);
    self_term_kernel<<<gNF, blk, 0, stream>>>(bufB, dinv, bufA, N);
    edge_scatter_kernel<<<gEdge, blk, 0, stream>>>(row, col, dinv, bufB, bufA, E);
    bn_relu_kernel<<<gNF, blk, 0, stream>>>(bufA, b1, bn1_g, bn1_b, bn1_m, bn1_v, bufB, N);

    // --- MLP head ---
    gemm_wmma_kernel<64, 2><<<gGemm, blk, 0, stream>>>(bufB, lin1_W, lin1_b, bufA, N);
    gemm_wmma_kernel<64, 2><<<gGemm, blk, 0, stream>>>(bufA, lin2_W, lin2_b, bufB, N);
    gemm_wmma_kernel<16, 1><<<gGemm, blk, 0, stream>>>(bufB, lin3_W, lin3_b, (float*)d_out, N);
}